// loss_generator_12163347383044
// MI455X (gfx1250) — compile-verified
//
#include <hip/hip_runtime.h>
#include <math.h>

// ---------------------------------------------------------------------------
// Fused fractional-PDE residual loss for MI455X (gfx1250, wave32).
//   - temporal frac contraction Du = M(11x11, padded 16x16) @ u via
//     V_WMMA_F32_16X16X4_F32 (full fp32 matrix pipe), K chunked by 4
//     (3 chunks cover K=0..11; valid K is 0..10, slot 11 zero-padded in LDS)
//   - halo staged global->LDS with CDNA5 async-LDS loads (ASYNCcnt path)
//   - 3x3 stencils (lap / px / py) read from LDS
//   - single-use forcing terms streamed with non-temporal loads
//   - wave shuffle reduction + one global float atomic per wave
// Memory bound: ~184MB @ 23.3 TB/s  => ~8us floor; all inputs read once.
// ---------------------------------------------------------------------------

typedef float v2f __attribute__((ext_vector_type(2)));
typedef float v8f __attribute__((ext_vector_type(8)));

#define T_REF   11
#define T_PAD   12              // extra zeroed time slot -> branch-free WMMA B
#define HW      1024
#define TILES_X 64              // ceil(1022/16)
#define ROWS    1022
#define WAVES_PER_BLOCK 8
#define NTASK   (ROWS * TILES_X)          // 65408 wave-tasks
#define NBLOCK  (NTASK / WAVES_PER_BLOCK) // 8176

// Padded 16x16 fractional-derivative matrix (row-major, pre-scaled by
// 0.1^-0.5 / gamma(1.5)). Rows 0 and 11..15 are zero; lower-triangular.
// Row i: [-s*w[i-1], s*(w[i-k]-w[i-1-k])..., s*w[0], 0...], w[j]=sqrt(j+1)-sqrt(j).
#define FD  3.5682483f
__constant__ float FRAC_TAB[256] = {
    // row 0
    0,0,0,0, 0,0,0,0, 0,0,0,0, 0,0,0,0,
    // row 1
    -3.5682483f, FD, 0,0, 0,0,0,0, 0,0,0,0, 0,0,0,0,
    // row 2
    -1.4780168f, -2.0902313f, FD, 0, 0,0,0,0, 0,0,0,0, 0,0,0,0,
    // row 3
    -1.1341221f, -0.3438946f, -2.0902313f, FD, 0,0,0,0, 0,0,0,0, 0,0,0,0,
    // row 4
    -0.9561092f, -0.1780130f, -0.3438946f, -2.0902313f, FD, 0,0,0, 0,0,0,0, 0,0,0,0,
    // row 5
    -0.8423501f, -0.1137601f, -0.1780130f, -0.3438946f, -2.0902313f, FD, 0,0,
    0,0,0,0, 0,0,0,0,
    // row 6
    -0.7615428f, -0.0808074f, -0.1137601f, -0.1780130f, -0.3438946f, -2.0902313f, FD, 0,
    0,0,0,0, 0,0,0,0,
    // row 7
    -0.7003110f, -0.0612319f, -0.0808074f, -0.1137601f, -0.1780130f, -0.3438946f,
    -2.0902313f, FD, 0,0,0,0, 0,0,0,0,
    // row 8
    -0.6518330f, -0.0484774f, -0.0612319f, -0.0808074f, -0.1137601f, -0.1780130f,
    -0.3438946f, -2.0902313f, FD, 0,0,0, 0,0,0,0,
    // row 9
    -0.6122146f, -0.0396180f, -0.0484774f, -0.0612319f, -0.0808074f, -0.1137601f,
    -0.1780130f, -0.3438946f, -2.0902313f, FD, 0,0, 0,0,0,0,
    // row 10
    -0.5790470f, -0.0331676f, -0.0396180f, -0.0484774f, -0.0612319f, -0.0808074f,
    -0.1137601f, -0.1780130f, -0.3438946f, -2.0902313f, FD, 0, 0,0,0,0,
    // rows 11..15
    0,0,0,0, 0,0,0,0, 0,0,0,0, 0,0,0,0,
    0,0,0,0, 0,0,0,0, 0,0,0,0, 0,0,0,0,
    0,0,0,0, 0,0,0,0, 0,0,0,0, 0,0,0,0,
    0,0,0,0, 0,0,0,0, 0,0,0,0, 0,0,0,0,
    0,0,0,0, 0,0,0,0, 0,0,0,0, 0,0,0,0
};

__global__ void zero_out_kernel(float* out) {
    if (threadIdx.x == 0 && blockIdx.x == 0) out[0] = 0.0f;
}

__global__ __launch_bounds__(256) void frac_pde_loss_kernel(
    const float* __restrict__ uv,   // (11, 2, 1024, 1024)
    const float* __restrict__ f1,   // (11, 1, 1024, 1024)
    const float* __restrict__ f2,   // (11, 1, 1024, 1024)
    float* __restrict__ loss)
{
    // per-wave halo tiles: [channel][t (11 + 1 zero pad)][3 rows][18 cols]
    __shared__ float lds[WAVES_PER_BLOCK][2][T_PAD][3][18];

    const int lane = threadIdx.x & 31;
    const int wv   = threadIdx.x >> 5;
    const int task = blockIdx.x * WAVES_PER_BLOCK + wv;
    const int y      = task / TILES_X + 1;        // 1..1022
    const int tile   = task % TILES_X;
    const int xstart = 1 + tile * 16;

    float (*lu)[3][18] = lds[wv][0];
    float (*lv)[3][18] = lds[wv][1];

    // Warm caches for the forcing terms we will need later.
    __builtin_prefetch(f1 + ((size_t)0 * HW + y) * HW + xstart, 0, 0);
    __builtin_prefetch(f2 + ((size_t)0 * HW + y) * HW + xstart, 0, 0);

    // Zero the padded time slot's center row (read by WMMA B for K=11).
    if (lane < 18) {
        lu[T_REF][1][lane] = 0.0f;
        lv[T_REF][1][lane] = 0.0f;
    }

    // ---- stage 11 x 3 x 18 halo of u and v into LDS via async-LDS loads ----
    for (int idx = lane; idx < T_REF * 3 * 18; idx += 32) {
        int t   = idx / 54;
        int rem = idx - t * 54;
        int dy  = rem / 18;
        int dxx = rem - dy * 18;
        int gx  = xstart - 1 + dxx; if (gx > 1023) gx = 1023;   // clamp last tile
        size_t base = (((size_t)t * 2) * HW + (size_t)(y - 1 + dy)) * HW + gx;
        unsigned long long gau = (unsigned long long)(uintptr_t)(uv + base);
        unsigned long long gav = gau + (unsigned long long)(HW * HW * sizeof(float));
        unsigned lou = (unsigned)(uintptr_t)(&lu[t][dy][dxx]);  // low 32b = LDS offset
        unsigned lov = (unsigned)(uintptr_t)(&lv[t][dy][dxx]);
        asm volatile("global_load_async_to_lds_b32 %0, %1, off"
                     :: "v"(lou), "v"(gau) : "memory");
        asm volatile("global_load_async_to_lds_b32 %0, %1, off"
                     :: "v"(lov), "v"(gav) : "memory");
    }
    asm volatile("s_wait_asynccnt 0x0" ::: "memory");
    __syncthreads();

    // ---- temporal contraction via fp32 WMMA:  D = M_pad(16x16) @ U(16x16) ----
    // A 16x4 layout: lanes 0-15 row M=lane, VGPR0=K0 / VGPR1=K1; lanes 16-31 K2/K3.
    // B 4x16: row K striped across lanes (N=lane%16), mirrored halves.
    // 3 K-chunks (K=0..11) fully cover the valid K=0..10; K=11 reads LDS zeros.
    const int mrow = lane & 15;
    const int koff = (lane < 16) ? 0 : 2;
    const int col  = lane & 15;

    v8f du = {};
    v8f dv = {};
#pragma unroll
    for (int kc = 0; kc < 3; ++kc) {
        int k0 = 4 * kc + koff;
        // unconditional 8-byte lookup of two adjacent A entries (8B aligned)
        v2f a = *(const v2f*)(&FRAC_TAB[mrow * 16 + k0]);
        v2f bu, bv;
        bu.x = lu[k0    ][1][1 + col];
        bu.y = lu[k0 + 1][1][1 + col];
        bv.x = lv[k0    ][1][1 + col];
        bv.y = lv[k0 + 1][1][1 + col];
        du = __builtin_amdgcn_wmma_f32_16x16x4_f32(false, a, false, bu,
                                                   (short)0, du, false, false);
        dv = __builtin_amdgcn_wmma_f32_16x16x4_f32(false, a, false, bv,
                                                   (short)0, dv, false, false);
    }

    // ---- stencils + residual in WMMA C/D layout: VGPR r -> t = r + 8*(lane/16)
    const float Lc = 0.34520446044393f;    // laplacian corner
    const float Le = 0.309591078922457f;   // laplacian edge
    const float Lm = -2.619182157203629f;  // laplacian center
    const float Pc = 0.086301115118584f;   // deriv corner
    const float Pe = 0.32739776970073f;    // deriv edge
    const float INV_DX2 = 1048576.0f;      // 1/DX^2
    const float INV_DX  = 1024.0f;         // 1/DX
    const float Rc = 0.01f;

    const int   thalf  = (lane >> 4) * 8;
    const int   px     = xstart + col;
    const bool  pvalid = (px <= 1022);
    float acc = 0.0f;

#pragma unroll
    for (int r = 0; r < 8; ++r) {
        int t = r + thalf;
        if (t < T_REF && pvalid) {
            float un[3][3], vn[3][3];
#pragma unroll
            for (int dy = 0; dy < 3; ++dy)
#pragma unroll
                for (int dxx = 0; dxx < 3; ++dxx) {
                    un[dy][dxx] = lu[t][dy][col + dxx];
                    vn[dy][dxx] = lv[t][dy][col + dxx];
                }

            float lap_u = (Lc * (un[0][0] + un[0][2] + un[2][0] + un[2][2]) +
                           Le * (un[0][1] + un[1][0] + un[1][2] + un[2][1]) +
                           Lm * un[1][1]) * INV_DX2;
            float u_x = (Pc * ((un[0][2] - un[0][0]) + (un[2][2] - un[2][0])) +
                         Pe * (un[1][2] - un[1][0])) * INV_DX;
            float u_y = (Pc * ((un[2][0] - un[0][0]) + (un[2][2] - un[0][2])) +
                         Pe * (un[2][1] - un[0][1])) * INV_DX;

            float lap_v = (Lc * (vn[0][0] + vn[0][2] + vn[2][0] + vn[2][2]) +
                           Le * (vn[0][1] + vn[1][0] + vn[1][2] + vn[2][1]) +
                           Lm * vn[1][1]) * INV_DX2;
            float v_x = (Pc * ((vn[0][2] - vn[0][0]) + (vn[2][2] - vn[2][0])) +
                         Pe * (vn[1][2] - vn[1][0])) * INV_DX;
            float v_y = (Pc * ((vn[2][0] - vn[0][0]) + (vn[2][2] - vn[0][2])) +
                         Pe * (vn[2][1] - vn[0][1])) * INV_DX;

            size_t fidx = ((size_t)t * HW + (size_t)y) * HW + px;
            float f1c = __builtin_nontemporal_load(f1 + fidx);  // single-use stream
            float f2c = __builtin_nontemporal_load(f2 + fidx);
            float uc = un[1][1], vc = vn[1][1];
            float fu = du[r] + uc * u_x + vc * u_y - Rc * lap_u - f1c;
            float fv = dv[r] + uc * v_x + vc * v_y - Rc * lap_v - f2c;
            acc = fmaf(fu, fu, acc);
            acc = fmaf(fv, fv, acc);
        }
    }

    // ---- wave32 reduction, one atomic per wave ----
#pragma unroll
    for (int off = 16; off > 0; off >>= 1)
        acc += __shfl_xor(acc, off, 32);

    const float INV_N = 1.0f / (11.0f * 1022.0f * 1022.0f); // mean(fu^2)+mean(fv^2)
    if (lane == 0)
        atomicAdd(loss, acc * INV_N);
}

extern "C" void kernel_launch(void* const* d_in, const int* in_sizes, int n_in,
                              void* d_out, int out_size, void* d_ws, size_t ws_size,
                              hipStream_t stream) {
    (void)in_sizes; (void)n_in; (void)out_size; (void)d_ws; (void)ws_size;
    const float* uv = (const float*)d_in[0];  // output: (11,2,1024,1024) fp32
    const float* f1 = (const float*)d_in[1];  // (11,1,1024,1024) fp32
    const float* f2 = (const float*)d_in[2];  // (11,1,1024,1024) fp32
    float* loss = (float*)d_out;              // scalar fp32

    zero_out_kernel<<<1, 64, 0, stream>>>(loss);
    frac_pde_loss_kernel<<<NBLOCK, 256, 0, stream>>>(uv, f1, f2, loss);
}